// ModelNew_28776280883297
// MI455X (gfx1250) — compile-verified
//
#include <hip/hip_runtime.h>
#include <hip/hip_bf16.h>

// C(16 x 16384) = A(16 x 16384) @ B(16384 x 16384), all fp32.
// Memory-bound: B is 1 GB read-once -> ~46 us floor at 23.3 TB/s.
// fp32 WMMA 16x16x4 (exact fp32 math); one wave per 2 adjacent 16x16 N-tiles
// (shared A fragment); K split 8 ways into workspace partials (compile-time
// trip count), then a tiny reduction kernel.

typedef __attribute__((ext_vector_type(2))) float v2f;
typedef __attribute__((ext_vector_type(4))) float v4f;
typedef __attribute__((ext_vector_type(8))) float v8f;

#define M_DIM 16
#define N_DIM 16384
#define K_DIM 16384
#define WAVES_PER_BLOCK 8     // 256 threads = 8 wave32
#define NTILES 2              // 16x16 N-tiles per wave (share one A fragment)
#define KSPLIT 8              // K-slices (needs KSPLIT MB of workspace)

static __device__ __forceinline__ v8f wmma_f32(v2f a, v2f b, v8f c) {
  return __builtin_amdgcn_wmma_f32_16x16x4_f32(
      /*neg_a=*/false, a, /*neg_b=*/false, b,
      /*c_mod=*/(short)0, c, /*reuse_a=*/false, /*reuse_b=*/false);
}

// Fragment layouts (CDNA5 ISA 7.12.2, wave32):
//  A 16x4 f32 : lane = M row; VGPR0 = K=0 (lanes 0-15) / K=2 (lanes 16-31),
//               VGPR1 = K=1 / K=3.  Row-major A => one b64 per lane.
//  B 4x16 f32 : lane = N col;  VGPR0 = K=0 / K=2, VGPR1 = K=1 / K=3.
//               Row-major B => two b32 loads, one k-row (N floats) apart.
//  C/D 16x16  : VGPR r = row r (lanes 0-15) / row r+8 (lanes 16-31).
template <int KSLICES>
__global__ __launch_bounds__(32 * WAVES_PER_BLOCK)
void skinny_gemm_wmma_f32(const float* __restrict__ A,
                          const float* __restrict__ B,
                          float* __restrict__ out) {  // [KSLICES][16][N]
  constexpr int K_PER_SLICE = K_DIM / KSLICES;  // compile-time trip count

  const int lane  = threadIdx.x & 31;
  const int wave  = threadIdx.x >> 5;
  const int n0    = (blockIdx.x * WAVES_PER_BLOCK + wave) * (16 * NTILES);
  const int slice = blockIdx.y;
  const int k0    = slice * K_PER_SLICE;

  const int sub   = lane & 15;          // M row (A) / N column offset (B)
  const int khalf = (lane >> 4) * 2;    // 0 for lanes 0-15, 2 for lanes 16-31

  const float* aptr = A + (size_t)sub * K_DIM + k0 + khalf;        // + k
  const float* bptr = B + (size_t)(k0 + khalf) * N_DIM + n0 + sub; // + k*N

  v8f acc0 = {};
  v8f acc1 = {};

#pragma unroll 8
  for (int k = 0; k < K_PER_SLICE; k += 4) {
    // Shared A fragment: one 8B load per lane (8B aligned: k,khalf,K all even)
    v2f afrag = *(const v2f*)(aptr + k);

    // B fragments for two adjacent N-tiles, non-temporal (stream-once).
    // Per k-row this wave touches 128 contiguous bytes.
    const float* bk = bptr + (size_t)k * N_DIM;
    v2f b0, b1;
    b0.x = __builtin_nontemporal_load(bk);
    b0.y = __builtin_nontemporal_load(bk + N_DIM);
    b1.x = __builtin_nontemporal_load(bk + 16);
    b1.y = __builtin_nontemporal_load(bk + N_DIM + 16);

    acc0 = wmma_f32(afrag, b0, acc0);
    acc1 = wmma_f32(afrag, b1, acc1);
  }

  // Store both 16x16 tiles: VGPR r -> row (r + 8*hi), col n0 + sub (+16)
  float* o = out + (size_t)slice * (M_DIM * (size_t)N_DIM);
  const int col   = n0 + sub;
  const int rbase = (lane >> 4) * 8;
#pragma unroll
  for (int r = 0; r < 8; ++r) {
    float* orow = o + (size_t)(rbase + r) * N_DIM;
    orow[col]      = acc0[r];
    orow[col + 16] = acc1[r];
  }
}

// Sum the KSPLIT partial C buffers into the final output. 4 floats/thread.
__global__ __launch_bounds__(256)
void reduce_ksplit(const float* __restrict__ ws, float* __restrict__ out,
                   int ksplit) {
  const size_t i = ((size_t)blockIdx.x * blockDim.x + threadIdx.x) * 4;
  v4f s = *(const v4f*)(ws + i);
  for (int p = 1; p < ksplit; ++p) {
    s += *(const v4f*)(ws + (size_t)p * (M_DIM * (size_t)N_DIM) + i);
  }
  *(v4f*)(out + i) = s;
}

extern "C" void kernel_launch(void* const* d_in, const int* in_sizes, int n_in,
                              void* d_out, int out_size, void* d_ws, size_t ws_size,
                              hipStream_t stream) {
  const float* A = (const float*)d_in[0];   // (16, 16384) fp32
  const float* B = (const float*)d_in[1];   // (16384, 16384) fp32
  float* out = (float*)d_out;               // (16, 16384) fp32

  const size_t part_elems = (size_t)M_DIM * N_DIM;
  const size_t ws_needed  = (size_t)KSPLIT * part_elems * sizeof(float);

  dim3 block(32 * WAVES_PER_BLOCK);
  const int gx = N_DIM / (16 * NTILES * WAVES_PER_BLOCK);  // 64

  // Deterministic: depends only on ws_size, which is fixed across calls.
  if (ws_size >= ws_needed) {
    dim3 grid(gx, KSPLIT);
    skinny_gemm_wmma_f32<KSPLIT>
        <<<grid, block, 0, stream>>>(A, B, (float*)d_ws);
    const int threads = 256;
    const int blocks  = (int)(part_elems / 4 / threads);  // 256
    reduce_ksplit<<<blocks, threads, 0, stream>>>((const float*)d_ws, out,
                                                  KSPLIT);
  } else {
    dim3 grid(gx, 1);
    skinny_gemm_wmma_f32<1><<<grid, block, 0, stream>>>(A, B, out);
  }
}